// GBNDecoder_53352083751388
// MI455X (gfx1250) — compile-verified
//
#include <hip/hip_runtime.h>
#include <math.h>

// Problem constants (fixed by the reference)
#define N_NODES 100000
#define DIM     128
#define NCLS    16
#define NSEED   16
#define NEDGE   3200000
#define KSEL    64
#define NITER   4
#define NTILES  (N_NODES / 16)   // 6250 node tiles of 16
#define ESTRIDE 132              // LDS row stride (132 % 64 == 4 -> conflict-free)

typedef __attribute__((ext_vector_type(2))) float v2f;
typedef __attribute__((ext_vector_type(8))) float v8f;

// ---------------------------------------------------------------- utilities
__device__ __forceinline__ unsigned ordKey(float f) {
    unsigned u = __float_as_uint(f);
    return (u & 0x80000000u) ? ~u : (u | 0x80000000u);   // order-preserving map
}
__device__ __forceinline__ float invOrd(unsigned u) {
    u = (u & 0x80000000u) ? (u ^ 0x80000000u) : ~u;
    return __uint_as_float(u);
}

// -------------------------------------------------- row 1/(||es||+1e-8)
__global__ __launch_bounds__(256) void k_row_norms(const float* __restrict__ es,
                                                   float* __restrict__ esInv) {
    int row  = blockIdx.x * 8 + (threadIdx.x >> 5);
    int lane = threadIdx.x & 31;
    if (row >= N_NODES) return;
    float4 v = ((const float4*)(es + (long)row * DIM))[lane];   // 32 lanes * 4 = 128
    float s = v.x * v.x + v.y * v.y + v.z * v.z + v.w * v.w;
    for (int off = 16; off > 0; off >>= 1) s += __shfl_down(s, off, 32);
    if (lane == 0) esInv[row] = 1.0f / (sqrtf(s) + 1e-8f);
}

__global__ void k_clear(unsigned* __restrict__ p, int n) {
    for (int i = blockIdx.x * blockDim.x + threadIdx.x; i < n;
         i += gridDim.x * blockDim.x) p[i] = 0u;
}

__global__ void k_seed_init(const int* __restrict__ seeds,
                            unsigned* __restrict__ cate,
                            unsigned* __restrict__ known) {
    int i = blockIdx.x * blockDim.x + threadIdx.x;
    if (i < NCLS * NSEED) {
        int c = i / NSEED;
        int node = seeds[i];
        atomicOr(&cate[node], 1u << c);
        atomicOr(&known[node], 1u);
    }
}

// ----------------------- per-edge: valid[dst] |= cateBits[src]  (MIN_MATCH==1)
__global__ __launch_bounds__(256) void k_edges(const int* __restrict__ src,
                                               const int* __restrict__ dst,
                                               const unsigned* __restrict__ cate,
                                               unsigned* __restrict__ valid) {
    for (int e = blockIdx.x * blockDim.x + threadIdx.x; e < NEDGE;
         e += gridDim.x * blockDim.x) {
        unsigned m = cate[src[e]];
        if (m) atomicOr(&valid[dst[e]], m);
    }
}

// ------------------------------------------------ memory step (1 block, 256 thr)
// hxNew = tanh((ctx [+ hxPrev]) @ W); ctx = softmax-attended read of es[last].
// The 16x128 @ 128x128 GEMM is done with V_WMMA_F32_16X16X4_F32 (8 waves x 32 k-steps).
__global__ __launch_bounds__(256) void k_memory_step(
    const float* __restrict__ es, const float* __restrict__ W,
    const int* __restrict__ last, int kcnt, int useHx,
    const float* __restrict__ hxPrev, float* __restrict__ hxOut,
    float* __restrict__ hxnOut, float* __restrict__ hxesOut) {
    __shared__ float qS[NCLS][DIM];
    __shared__ float aS[NCLS][KSEL];
    __shared__ __align__(16) float preS[NCLS][ESTRIDE];
    __shared__ float hxS[NCLS][DIM];
    __shared__ float invS[NCLS];
    __shared__ int   lastS[NCLS * KSEL];
    int tid = threadIdx.x;

    for (int i = tid; i < NCLS * kcnt; i += 256) lastS[i] = last[i];
    __syncthreads();

    // q = mean(inp) on iter 0, else hx
    for (int i = tid; i < NCLS * DIM; i += 256) {
        int c = i / DIM, d = i % DIM;
        float q;
        if (useHx) q = hxPrev[i];
        else {
            float s = 0.f;
            for (int j = 0; j < kcnt; ++j) s += es[(long)lastS[c * kcnt + j] * DIM + d];
            q = s / (float)kcnt;
        }
        qS[c][d] = q;
    }
    __syncthreads();

    // logits = inp . q / sqrt(D)
    for (int p = tid; p < NCLS * kcnt; p += 256) {
        int c = p / kcnt, j = p % kcnt;
        const float* row = es + (long)lastS[c * kcnt + j] * DIM;
        float s = 0.f;
        for (int d = 0; d < DIM; ++d) s += row[d] * qS[c][d];
        aS[c][j] = s * 0.08838834764831845f;   // 1/sqrt(128)
    }
    __syncthreads();

    // softmax per class
    if (tid < NCLS) {
        int c = tid;
        float mx = -INFINITY;
        for (int j = 0; j < kcnt; ++j) mx = fmaxf(mx, aS[c][j]);
        float sum = 0.f;
        for (int j = 0; j < kcnt; ++j) { float e = __expf(aS[c][j] - mx); aS[c][j] = e; sum += e; }
        float inv = 1.f / sum;
        for (int j = 0; j < kcnt; ++j) aS[c][j] *= inv;
    }
    __syncthreads();

    // pre = attended ctx (+ hxPrev)
    for (int i = tid; i < NCLS * DIM; i += 256) {
        int c = i / DIM, d = i % DIM;
        float s = 0.f;
        for (int j = 0; j < kcnt; ++j) s += aS[c][j] * es[(long)lastS[c * kcnt + j] * DIM + d];
        if (useHx) s += hxPrev[i];
        preS[c][d] = s;
    }
    __syncthreads();

    // WMMA: D[16 classes x 16 cols] per wave; 8 waves cover 128 output cols.
    int lane = tid & 31;
    int n = lane & 15, hi = lane >> 4;
    int colbase = (tid >> 5) * 16;
    v8f acc = {};
    for (int k0 = 0; k0 < DIM; k0 += 4) {
        v2f a, b;
        a.x = preS[n][k0 + 2 * hi];                      // A: M=lane%16, K=k0+v+2*hi
        a.y = preS[n][k0 + 2 * hi + 1];
        b.x = W[(long)(k0 + 2 * hi) * DIM + colbase + n];     // B: K rows of W, N=cols
        b.y = W[(long)(k0 + 2 * hi + 1) * DIM + colbase + n];
        acc = __builtin_amdgcn_wmma_f32_16x16x4_f32(false, a, false, b,
                                                    (short)0, acc, false, false);
    }
#pragma unroll
    for (int v = 0; v < 8; ++v) {
        int c = v + 8 * hi;                               // D: M = v + 8*hi (class)
        hxS[c][colbase + n] = tanhf(acc[v]);
    }
    __syncthreads();

    if (tid < NCLS) {
        float s = 0.f;
        for (int d = 0; d < DIM; ++d) s += hxS[tid][d] * hxS[tid][d];
        invS[tid] = 1.f / (sqrtf(s) + 1e-8f);
    }
    __syncthreads();
    for (int i = tid; i < NCLS * DIM; i += 256) {
        int c = i / DIM;
        float h = hxS[c][i % DIM];
        hxOut[i] = h;
        hxesOut[i] = h;                 // output slot for this iteration
        hxnOut[i] = h * invS[c];        // normalized rows for the cosine GEMM
    }
}

// ------------------------------------ score GEMM: scores[c][n] = cosine(es_n, hx_c)
// 4 waves/block, one 16-node tile per wave. es tile staged via LDS (coalesced
// float4 loads), then 32 x V_WMMA_F32_16X16X4_F32 per tile. Post-scale by
// esInv and mask with (valid bit && !known) -> -1e9.
__global__ __launch_bounds__(128) void k_scores(
    const float* __restrict__ es, const float* __restrict__ hxn,
    const float* __restrict__ esInv, const unsigned* __restrict__ valid,
    const unsigned* __restrict__ known, float* __restrict__ scores) {
    __shared__ __align__(16) float esT[4][16][ESTRIDE];
    __shared__ __align__(16) float hxT[16][ESTRIDE];
    int tid = threadIdx.x, wv = tid >> 5, lane = tid & 31;

    for (int i = tid; i < NCLS * DIM; i += 128) hxT[i / DIM][i % DIM] = hxn[i];

    int tile = blockIdx.x * 4 + wv;
    int tC = tile < NTILES ? tile : NTILES - 1;           // clamp; store guarded below
    int nb = tC * 16;
    for (int r = 0; r < 16; ++r) {
        float4 vv = ((const float4*)(es + (long)(nb + r) * DIM))[lane];
        ((float4*)&esT[wv][r][0])[lane] = vv;             // 528B row stride, 16B aligned
    }
    __syncthreads();

    int n = lane & 15, hi = lane >> 4;
    v8f acc = {};
    for (int k0 = 0; k0 < DIM; k0 += 4) {
        v2f a, b;
        a.x = esT[wv][n][k0 + 2 * hi];                    // A: M=node row, K
        a.y = esT[wv][n][k0 + 2 * hi + 1];
        b.x = hxT[n][k0 + 2 * hi];                        // B: K x N, N=class
        b.y = hxT[n][k0 + 2 * hi + 1];
        acc = __builtin_amdgcn_wmma_f32_16x16x4_f32(false, a, false, b,
                                                    (short)0, acc, false, false);
    }
    if (tile < NTILES) {
        int cls = n;
#pragma unroll
        for (int v = 0; v < 8; ++v) {
            int node = nb + v + 8 * hi;                   // D: M = v + 8*hi (node row)
            float sc = acc[v] * esInv[node];              // hx already normalized
            bool ok = ((valid[node] >> cls) & 1u) && (known[node] == 0u);
            scores[(long)cls * N_NODES + node] = ok ? sc : -1e9f;
        }
    }
}

// --------------------------- per-class top-K: histogram threshold + bitonic sort
#define NBIN 2048
#define CAP  4096
__global__ __launch_bounds__(256) void k_select(
    const float* __restrict__ scores, unsigned* __restrict__ cate,
    unsigned* __restrict__ known, int* __restrict__ lastIdx,
    float* __restrict__ outsProbs, float* __restrict__ outsIdx) {
    __shared__ unsigned hist[NBIN];
    __shared__ unsigned long long keys[CAP];
    __shared__ unsigned thrBin, cnt;
    int c = blockIdx.x, tid = threadIdx.x;
    const float* sc = scores + (long)c * N_NODES;

    for (int i = tid; i < NBIN; i += 256) hist[i] = 0u;
    if (tid == 0) cnt = 0u;
    __syncthreads();

    for (int i = tid; i < N_NODES; i += 256)
        atomicAdd(&hist[ordKey(sc[i]) >> 21], 1u);
    __syncthreads();

    if (tid == 0) {
        unsigned cum = 0; int b = NBIN - 1;
        for (; b > 0; --b) { cum += hist[b]; if (cum >= KSEL) break; }
        if (cum < KSEL) b = 0;
        thrBin = (unsigned)b;
    }
    __syncthreads();

    unsigned tb = thrBin;
    for (int i = tid; i < N_NODES; i += 256) {
        unsigned u = ordKey(sc[i]);
        if ((u >> 21) >= tb) {
            unsigned pos = atomicAdd(&cnt, 1u);
            if (pos < CAP)
                keys[pos] = ((unsigned long long)u << 32) | (unsigned)(~i);
        }
    }
    __syncthreads();
    unsigned m = cnt < CAP ? cnt : (unsigned)CAP;
    for (int i = tid; i < CAP; i += 256)
        if ((unsigned)i >= m) keys[i] = 0ull;

    // bitonic sort, descending: key = (ordered score << 32) | ~index
    for (unsigned k = 2; k <= CAP; k <<= 1) {
        for (unsigned j = k >> 1; j > 0; j >>= 1) {
            __syncthreads();
            for (unsigned i = tid; i < CAP; i += 256) {
                unsigned l = i ^ j;
                if (l > i) {
                    unsigned long long a = keys[i], b = keys[l];
                    bool desc = ((i & k) == 0);
                    if (desc ? (a < b) : (a > b)) { keys[i] = b; keys[l] = a; }
                }
            }
        }
    }
    __syncthreads();

    if (tid < KSEL) {
        unsigned long long key = keys[tid];
        unsigned u = (unsigned)(key >> 32);
        int node = (int)(~(unsigned)key);
        outsProbs[c * KSEL + tid] = invOrd(u);
        outsIdx[c * KSEL + tid] = (float)node;
        lastIdx[c * KSEL + tid] = node;
        atomicOr(&cate[node], 1u << c);
        atomicOr(&known[node], 1u);
    }
}

// ------------------------------------------------------------------ launcher
extern "C" void kernel_launch(void* const* d_in, const int* in_sizes, int n_in,
                              void* d_out, int out_size, void* d_ws, size_t ws_size,
                              hipStream_t stream) {
    (void)in_sizes; (void)n_in; (void)out_size; (void)ws_size;
    const float* es    = (const float*)d_in[0];
    const int*   edge  = (const int*)d_in[1];
    const int*   seeds = (const int*)d_in[2];
    const float* W     = (const float*)d_in[3];
    // d_in[4] = n_iter, fixed at 4 by the reference constants
    const int* src = edge;
    const int* dst = edge + NEDGE;

    char* ws = (char*)d_ws;
    size_t off = 0;
    auto alloc = [&](size_t bytes) -> void* {
        void* p = ws + off;
        off += (bytes + 255) & ~(size_t)255;
        return p;
    };
    float*    esInv   = (float*)alloc((size_t)N_NODES * 4);
    unsigned* cate    = (unsigned*)alloc((size_t)N_NODES * 4);
    unsigned* known   = (unsigned*)alloc((size_t)N_NODES * 4);
    unsigned* valid   = (unsigned*)alloc((size_t)N_NODES * 4);
    float*    hxA     = (float*)alloc(NCLS * DIM * 4);
    float*    hxB     = (float*)alloc(NCLS * DIM * 4);
    float*    hxn     = (float*)alloc(NCLS * DIM * 4);
    int*      lastIdx = (int*)alloc(NCLS * KSEL * 4);
    float*    scores  = (float*)alloc((size_t)NCLS * N_NODES * 4);

    float* outs = (float*)d_out;                  // [4][16][64]
    float* exps = outs + NITER * NCLS * KSEL;     // [4][16][64] (indices as float)
    float* hxes = exps + NITER * NCLS * KSEL;     // [4][16][128]

    k_row_norms<<<(N_NODES + 7) / 8, 256, 0, stream>>>(es, esInv);
    k_clear<<<256, 256, 0, stream>>>(cate, N_NODES);
    k_clear<<<256, 256, 0, stream>>>(known, N_NODES);
    k_seed_init<<<1, 256, 0, stream>>>(seeds, cate, known);

    const float* hxPrev = nullptr;
    float* hxCur = hxA;
    for (int t = 0; t < NITER; ++t) {
        int kcnt = (t == 0) ? NSEED : KSEL;
        const int* last = (t == 0) ? seeds : lastIdx;
        k_memory_step<<<1, 256, 0, stream>>>(es, W, last, kcnt, t > 0, hxPrev,
                                             hxCur, hxn, hxes + t * NCLS * DIM);
        k_clear<<<256, 256, 0, stream>>>(valid, N_NODES);
        k_edges<<<2048, 256, 0, stream>>>(src, dst, cate, valid);
        k_scores<<<(NTILES + 3) / 4, 128, 0, stream>>>(es, hxn, esInv, valid,
                                                       known, scores);
        k_select<<<NCLS, 256, 0, stream>>>(scores, cate, known, lastIdx,
                                           outs + t * NCLS * KSEL,
                                           exps + t * NCLS * KSEL);
        hxPrev = hxCur;
        hxCur = (hxCur == hxA) ? hxB : hxA;
    }
}